// NATTENLayer_26053271617844
// MI455X (gfx1250) — compile-verified
//
#include <hip/hip_runtime.h>
#include <hip/hip_bf16.h>
#include <math.h>

// ---------------------------------------------------------------------------
// NATTEN layer for MI455X (gfx1250, wave32, WMMA + async global->LDS).
// fp32 end-to-end using V_WMMA_F32_16X16X4_F32 (exact fp32 matrix path).
// ---------------------------------------------------------------------------

typedef __attribute__((ext_vector_type(2))) float v2f;
typedef __attribute__((ext_vector_type(8))) float v8f;

#define DIMX    512
#define NHEADS  8
#define HDIM    64
#define L_SEQ   2048
#define BATCH   2
#define M_ROWS  (BATCH * L_SEQ)   /* 4096 */
#define HALF_W  15
#define LN_EPS  1e-5f
#define TPAD    516               /* LDS row stride (floats): 516 % 64 == 4 */

__device__ __forceinline__ v8f wmma_f32(v2f a, v2f b, v8f c) {
  // (neg_a, A, neg_b, B, c_mod, C, reuse_a, reuse_b)
  return __builtin_amdgcn_wmma_f32_16x16x4_f32(false, a, false, b, (short)0, c,
                                               false, false);
}

// Async copy of one 16-byte packet global -> LDS (CDNA5 async engine,
// tracked by ASYNCcnt; no VGPR round trip).
__device__ __forceinline__ void async_b128(unsigned lds_off,
                                           const void* gaddr) {
  asm volatile("global_load_async_to_lds_b128 %0, %1, off"
               :
               : "v"(lds_off), "v"(gaddr)
               : "memory");
}

__device__ __forceinline__ void async_wait0() {
  asm volatile("s_wait_asynccnt 0x0" ::: "memory");
}

// Stage a 16 x 512 fp32 tile (row stride DIMX in global) into LDS with row
// stride TPAD, using 8 async b128 copies per thread (256 threads).
__device__ __forceinline__ void stage_tile_async(const float* __restrict__ src,
                                                 float* lds_tile, int tid) {
  const unsigned base = (unsigned)(size_t)lds_tile;  // low 32 bits = LDS addr
#pragma unroll
  for (int it = 0; it < 8; ++it) {
    const int q    = it * 256 + tid;   // 0..2047 b128 packets
    const int row  = q >> 7;           // 0..15
    const int col4 = q & 127;          // 0..127 (packet of 4 floats)
    const float* g = src + (size_t)row * DIMX + col4 * 4;
    const unsigned l = base + (unsigned)(row * TPAD + col4 * 4) * 4u;
    async_b128(l, g);
  }
  async_wait0();
}

// ---------------------------------------------------------------------------
// Kernel 1: QKV GEMM.  C(4096x1536) = X(4096x512) @ Wqkv(512x1536) + bqkv.
// Block = one 16-row M-tile x one 512-col group (exactly one of Q/K/V).
// The 16x512 X tile is async-staged to LDS once and shared by all 8 waves;
// wave w covers head w's 64 columns.
// ---------------------------------------------------------------------------
__global__ __launch_bounds__(256) void qkv_gemm_kernel(
    const float* __restrict__ X, const float* __restrict__ W,
    const float* __restrict__ bias,
    float* __restrict__ Q, float* __restrict__ K, float* __restrict__ V) {
  __shared__ float tile[16 * TPAD];

  const int tid  = threadIdx.x;
  const int lane = tid & 31;
  const int wave = tid >> 5;
  const int mt   = blockIdx.x / 3;  // 0..255 (16-row tile)
  const int cg   = blockIdx.x % 3;  // 0=Q, 1=K, 2=V column group
  const int half = lane >> 4;
  const int ln   = lane & 15;
  const int n0   = cg * DIMX + wave * 64;  // global column base

  stage_tile_async(X + (size_t)mt * 16 * DIMX, tile, tid);
  __syncthreads();

  const float* arow = tile + ln * TPAD;  // A row for this lane (LDS)

  v8f acc[4] = {};
  for (int k0 = 0; k0 < DIMX; k0 += 4) {
    const int k = k0 + 2 * half;
    v2f a = *(const v2f*)(arow + k);
    const float* w0 = W + (size_t)k * (3 * DIMX) + n0 + ln;
    const float* w1 = w0 + 3 * DIMX;
#pragma unroll
    for (int t = 0; t < 4; ++t) {
      v2f b;
      b.x = w0[t * 16];
      b.y = w1[t * 16];
      acc[t] = wmma_f32(a, b, acc[t]);
    }
  }

  // Scatter into (B, H, L, 64); cg selects the tensor, wave == head.
  float* base = (cg == 0) ? Q : ((cg == 1) ? K : V);
  const int h = wave;
#pragma unroll
  for (int t = 0; t < 4; ++t) {
    const int d    = wave * 64 + t * 16 + ln - h * 64;  // == t*16+ln
    const float bv = bias[n0 + t * 16 + ln];
#pragma unroll
    for (int r = 0; r < 8; ++r) {
      const int row  = mt * 16 + r + 8 * half;  // 0..4095
      const int bidx = row >> 11;
      const int l    = row & (L_SEQ - 1);
      base[(((size_t)bidx * NHEADS + h) * L_SEQ + l) * HDIM + d] =
          acc[t][r] + bv;
    }
  }
}

// ---------------------------------------------------------------------------
// Kernel 2: banded attention.  One wave per (b,h,16-query tile).
// S = Q Kt over 3 aligned key tiles (48 keys), mask |i-j|<=15, softmax,
// O = P V.  P is staged through padded LDS to convert C-layout -> A-layout.
// ---------------------------------------------------------------------------
__global__ __launch_bounds__(256) void natten_attn_kernel(
    const float* __restrict__ Q, const float* __restrict__ K,
    const float* __restrict__ V, float* __restrict__ AO) {
  __shared__ float lds_p[8][16][49];  // [wave][row][key], padded stride

  const int lane = threadIdx.x & 31;
  const int wave = threadIdx.x >> 5;
  const int gw   = blockIdx.x * 8 + wave;  // 0..2047
  const int bh   = gw >> 7;                // 0..15
  const int qt   = gw & 127;               // query tile
  const int half = lane >> 4;
  const int ln   = lane & 15;

  const size_t hoff = (size_t)bh * L_SEQ * HDIM;
  const float* qrow = Q + hoff + (size_t)(qt * 16 + ln) * HDIM;
  const float* kbas = K + hoff;
  const float* vbas = V + hoff;

  // ---- S = Q K^T (scale later), 3 key tiles of 16 ----
  v8f s[3] = {};
#pragma unroll
  for (int t = 0; t < 3; ++t) {
    const int ktile = qt - 1 + t;
    if (ktile < 0 || ktile >= (L_SEQ / 16)) continue;  // wave-uniform
    const float* krow = kbas + (size_t)(ktile * 16 + ln) * HDIM;
    v8f a8 = {};
#pragma unroll
    for (int k0 = 0; k0 < HDIM; k0 += 4) {
      v2f a = *(const v2f*)(qrow + k0 + 2 * half);
      v2f b = *(const v2f*)(krow + k0 + 2 * half);
      a8 = wmma_f32(a, b, a8);
    }
    s[t] = a8;
  }

  // ---- masked softmax, row-wise (rows live on 16-lane halves) ----
  const float scale = 0.125f;  // 1/sqrt(64)
#pragma unroll
  for (int r = 0; r < 8; ++r) {
    const int i = qt * 16 + r + 8 * half;
    float sv[3];
    float m = -1e30f;
#pragma unroll
    for (int t = 0; t < 3; ++t) {
      const int j = (qt - 1 + t) * 16 + ln;
      const bool valid =
          (j >= 0) && (j < L_SEQ) && (i - j <= HALF_W) && (j - i <= HALF_W);
      sv[t] = valid ? s[t][r] * scale : -1e30f;
      m = fmaxf(m, sv[t]);
    }
#pragma unroll
    for (int off = 1; off < 16; off <<= 1) m = fmaxf(m, __shfl_xor(m, off, 32));
    float sum = 0.f;
#pragma unroll
    for (int t = 0; t < 3; ++t) {
      sv[t] = __expf(sv[t] - m);
      sum += sv[t];
    }
#pragma unroll
    for (int off = 1; off < 16; off <<= 1) sum += __shfl_xor(sum, off, 32);
    const float inv = 1.0f / sum;  // diagonal always valid -> sum > 0
#pragma unroll
    for (int t = 0; t < 3; ++t)
      lds_p[wave][r + 8 * half][t * 16 + ln] = sv[t] * inv;
  }

  // ---- O = P V, K-dim = 48 ----
  v8f o[4] = {};
#pragma unroll
  for (int k0 = 0; k0 < 48; k0 += 4) {
    const int kk = k0 + 2 * half;
    v2f a;
    a.x = lds_p[wave][ln][kk];
    a.y = lds_p[wave][ln][kk + 1];
    int row0 = (qt - 1) * 16 + kk;  // key row; may be OOB where P == 0
    int row1 = row0 + 1;
    row0 = min(max(row0, 0), L_SEQ - 1);
    row1 = min(max(row1, 0), L_SEQ - 1);
#pragma unroll
    for (int t = 0; t < 4; ++t) {
      v2f b;
      b.x = vbas[(size_t)row0 * HDIM + t * 16 + ln];
      b.y = vbas[(size_t)row1 * HDIM + t * 16 + ln];
      o[t] = wmma_f32(a, b, o[t]);
    }
  }

  // ---- write attention output as (B, L, D) with col = h*64 + d ----
  const int bidx = bh >> 3;
  const int h    = bh & 7;
#pragma unroll
  for (int t = 0; t < 4; ++t) {
#pragma unroll
    for (int r = 0; r < 8; ++r) {
      const int l = qt * 16 + r + 8 * half;
      AO[((size_t)(bidx * L_SEQ + l)) * DIMX + h * HDIM + t * 16 + ln] =
          o[t][r];
    }
  }
}

// ---------------------------------------------------------------------------
// Kernel 3: fused proj GEMM + bias + residual + LayerNorm.
// One block per 16-row tile; AO tile async-staged to LDS and shared by all
// 8 waves (each owns a 64-col strip); LN via shfl + LDS cross-wave reduce.
// ---------------------------------------------------------------------------
__global__ __launch_bounds__(256) void proj_ln_kernel(
    const float* __restrict__ AO, const float* __restrict__ W,
    const float* __restrict__ bias, const float* __restrict__ X,
    const float* __restrict__ gamma, const float* __restrict__ beta,
    float* __restrict__ OUT) {
  __shared__ float tile[16 * TPAD];
  __shared__ float red_s[8][16];
  __shared__ float red_q[8][16];

  const int tid  = threadIdx.x;
  const int lane = tid & 31;
  const int wave = tid >> 5;
  const int mt   = blockIdx.x;  // 0..255
  const int half = lane >> 4;
  const int ln   = lane & 15;
  const int n0   = wave * 64;

  stage_tile_async(AO + (size_t)mt * 16 * DIMX, tile, tid);
  __syncthreads();

  const float* arow = tile + ln * TPAD;

  v8f acc[4] = {};
  for (int k0 = 0; k0 < DIMX; k0 += 4) {
    const int k = k0 + 2 * half;
    v2f a = *(const v2f*)(arow + k);
    const float* w0 = W + (size_t)k * DIMX + n0 + ln;
    const float* w1 = w0 + DIMX;
#pragma unroll
    for (int t = 0; t < 4; ++t) {
      v2f b;
      b.x = w0[t * 16];
      b.y = w1[t * 16];
      acc[t] = wmma_f32(a, b, acc[t]);
    }
  }

  // y = proj + bias + residual; accumulate per-row sum / sumsq
  float y[4][8];
  float psum[8], psq[8];
#pragma unroll
  for (int r = 0; r < 8; ++r) { psum[r] = 0.f; psq[r] = 0.f; }
#pragma unroll
  for (int t = 0; t < 4; ++t) {
    const int col  = n0 + t * 16 + ln;
    const float bv = bias[col];
#pragma unroll
    for (int r = 0; r < 8; ++r) {
      const int row = mt * 16 + r + 8 * half;
      const float v = acc[t][r] + bv + X[(size_t)row * DIMX + col];
      y[t][r] = v;
      psum[r] += v;
      psq[r] += v * v;
    }
  }
#pragma unroll
  for (int r = 0; r < 8; ++r) {
    float s = psum[r], q = psq[r];
#pragma unroll
    for (int off = 1; off < 16; off <<= 1) {
      s += __shfl_xor(s, off, 32);
      q += __shfl_xor(q, off, 32);
    }
    if (ln == 0) {
      red_s[wave][r + 8 * half] = s;
      red_q[wave][r + 8 * half] = q;
    }
  }
  __syncthreads();

  float mu[8], rstd[8];
#pragma unroll
  for (int r = 0; r < 8; ++r) {
    const int rowi = r + 8 * half;
    float s = 0.f, q = 0.f;
#pragma unroll
    for (int wv = 0; wv < 8; ++wv) {
      s += red_s[wv][rowi];
      q += red_q[wv][rowi];
    }
    const float m   = s * (1.0f / DIMX);
    const float var = q * (1.0f / DIMX) - m * m;
    mu[r]   = m;
    rstd[r] = rsqrtf(var + LN_EPS);
  }

#pragma unroll
  for (int t = 0; t < 4; ++t) {
    const int col  = n0 + t * 16 + ln;
    const float g  = gamma[col];
    const float bb = beta[col];
#pragma unroll
    for (int r = 0; r < 8; ++r) {
      const int row = mt * 16 + r + 8 * half;
      OUT[(size_t)row * DIMX + col] = (y[t][r] - mu[r]) * rstd[r] * g + bb;
    }
  }
}

// ---------------------------------------------------------------------------
extern "C" void kernel_launch(void* const* d_in, const int* in_sizes, int n_in,
                              void* d_out, int out_size, void* d_ws,
                              size_t ws_size, hipStream_t stream) {
  const float* x     = (const float*)d_in[0];
  const float* Wqkv  = (const float*)d_in[1];
  const float* bqkv  = (const float*)d_in[2];
  const float* Wproj = (const float*)d_in[3];
  const float* bproj = (const float*)d_in[4];
  const float* gamma = (const float*)d_in[5];
  const float* beta  = (const float*)d_in[6];
  float* out = (float*)d_out;

  float* ws = (float*)d_ws;
  const size_t per = (size_t)M_ROWS * DIMX;  // 4096*512 floats = 8 MB
  float* Q  = ws;
  float* K  = ws + per;
  float* V  = ws + 2 * per;
  float* AO = ws + 3 * per;

  // 256 M-tiles x 3 column groups (Q/K/V), 8 waves = 8 heads per block
  qkv_gemm_kernel<<<768, 256, 0, stream>>>(x, Wqkv, bqkv, Q, K, V);
  // 2 * 8 * 128 = 2048 query tiles, one wave each
  natten_attn_kernel<<<256, 256, 0, stream>>>(Q, K, V, AO);
  // one block per 16-row tile
  proj_ln_kernel<<<256, 256, 0, stream>>>(AO, Wproj, bproj, x, gamma, beta,
                                          out);
}